// HydraAttention_62749472194590
// MI455X (gfx1250) — compile-verified
//
#include <hip/hip_runtime.h>
#include <hip/hip_bf16.h>

typedef __attribute__((ext_vector_type(2))) float v2f;
typedef __attribute__((ext_vector_type(8))) float v8f;

#define B_   8
#define N_   4096
#define DIM_ 768
#define H_   12
#define DH_  64
#define G_   4
#define INNER_ 768

__device__ __forceinline__ v8f wmma4(v2f a, v2f b, v8f c) {
    // V_WMMA_F32_16X16X4_F32 : D(16x16,f32) = A(16x4,f32) * B(4x16,f32) + C
    return __builtin_amdgcn_wmma_f32_16x16x4_f32(false, a, false, b, (short)0, c, false, false);
}

__device__ __forceinline__ float geluf(float z) {
    return 0.5f * z * (1.0f + erff(z * 0.70710678118654752440f));
}

// ---------------------------------------------------------------------------
// Generic row-major GEMM:  C[M,N] = A[M,K] * B[K,N] (+ bias[N])
// Block: 256 thr (8 waves). Tile: 128 rows x 64 cols. K chunk = 16.
// Wave w owns rows [w*16, w*16+16), 4 accumulators of 16x16.
// ---------------------------------------------------------------------------
__global__ __launch_bounds__(256) void gemm_f32_wmma(
    const float* __restrict__ A, const float* __restrict__ Bm,
    const float* __restrict__ bias, float* __restrict__ C,
    int M, int N, int K) {
    __shared__ float lA[128 * 16];
    __shared__ float lB[16 * 64];
    const int t = threadIdx.x;
    const int lane = t & 31;
    const int wave = t >> 5;
    const int rowB = blockIdx.y * 128;
    const int colB = blockIdx.x * 64;

    v8f acc[4] = {};
    const int m   = wave * 16 + (lane & 15);
    const int koff = (lane >> 4) * 2;
    const int nn   = lane & 15;

    for (int k0 = 0; k0 < K; k0 += 16) {
        // stage A: 128x16
        {
            int idx = t * 4;
            int r = idx >> 4, c = idx & 15;
            *(float4*)(lA + r * 16 + c) =
                *(const float4*)(A + (size_t)(rowB + r) * K + k0 + c);
            idx += 1024; r = idx >> 4; c = idx & 15;
            *(float4*)(lA + r * 16 + c) =
                *(const float4*)(A + (size_t)(rowB + r) * K + k0 + c);
        }
        // stage B: 16x64
        {
            int idx = t * 4;
            int r = idx >> 6, c = idx & 63;
            *(float4*)(lB + r * 64 + c) =
                *(const float4*)(Bm + (size_t)(k0 + r) * N + colB + c);
        }
        __syncthreads();
        #pragma unroll
        for (int ks = 0; ks < 4; ks++) {
            const int kk = ks * 4 + koff;
            v2f a;
            a.x = lA[m * 16 + kk];
            a.y = lA[m * 16 + kk + 1];
            #pragma unroll
            for (int c = 0; c < 4; c++) {
                v2f b;
                b.x = lB[kk * 64 + c * 16 + nn];
                b.y = lB[(kk + 1) * 64 + c * 16 + nn];
                acc[c] = wmma4(a, b, acc[c]);
            }
        }
        __syncthreads();
    }
    const int half = lane >> 4;
    #pragma unroll
    for (int c = 0; c < 4; c++) {
        const int col = colB + c * 16 + nn;
        const float bv = bias ? bias[col] : 0.0f;
        #pragma unroll
        for (int j = 0; j < 8; j++) {
            const int row = rowB + wave * 16 + j + half * 8;
            C[(size_t)row * N + col] = acc[c][j] + bv;
        }
    }
}

// ---------------------------------------------------------------------------
// Per-head transform: q <- LN(gelu(q @ Wt[g] + bt[g])) * gt + betat, in place.
// One wave per (g,b,n,h) row of 64. Block = 256 thr = 8 rows, single g/block.
// ---------------------------------------------------------------------------
__global__ __launch_bounds__(256) void qtransform_kernel(
    float* __restrict__ q, const float* __restrict__ Wt,
    const float* __restrict__ bt, const float* __restrict__ gt,
    const float* __restrict__ betat) {
    __shared__ float lW[64 * 64];
    __shared__ float lb[64], lg[64], lbe[64];
    const long long rowBase = (long long)blockIdx.x * 8;
    const long long rowsPerG = (long long)B_ * N_ * H_;   // 393216
    const int g = (int)(rowBase / rowsPerG);

    for (int i = threadIdx.x; i < 64 * 64; i += 256) lW[i] = Wt[g * 4096 + i];
    if (threadIdx.x < 64) {
        lb[threadIdx.x]  = bt[g * 64 + threadIdx.x];
        lg[threadIdx.x]  = gt[g * 64 + threadIdx.x];
        lbe[threadIdx.x] = betat[g * 64 + threadIdx.x];
    }
    __syncthreads();

    const int wave = threadIdx.x >> 5, lane = threadIdx.x & 31;
    const long long row = rowBase + wave;                 // ((g*B+b)*N+n)*H + h
    float* base = q + (row / H_) * (size_t)INNER_ + (row % H_) * DH_;

    float x0 = base[lane], x1 = base[lane + 32];
    float o0 = lb[lane], o1 = lb[lane + 32];
    #pragma unroll 8
    for (int d = 0; d < 64; d++) {
        float src = (d < 32) ? x0 : x1;
        float qd = __shfl(src, d & 31, 32);
        o0 += qd * lW[d * 64 + lane];
        o1 += qd * lW[d * 64 + lane + 32];
    }
    o0 = geluf(o0);
    o1 = geluf(o1);
    float s = o0 + o1, s2 = o0 * o0 + o1 * o1;
    #pragma unroll
    for (int off = 16; off > 0; off >>= 1) {
        s  += __shfl_xor(s, off, 32);
        s2 += __shfl_xor(s2, off, 32);
    }
    const float mean = s * (1.0f / 64.0f);
    const float var  = s2 * (1.0f / 64.0f) - mean * mean;
    const float inv  = rsqrtf(var + 1e-5f);
    base[lane]      = (o0 - mean) * inv * lg[lane] + lbe[lane];
    base[lane + 32] = (o1 - mean) * inv * lg[lane + 32] + lbe[lane + 32];
}

// ---------------------------------------------------------------------------
// kv_kernel: per (b,h) compute kvx[128][80] = phi(k)^T @ [v | 1 | 0...],
// reduction over n=4096 via WMMA f32. Col 64 accumulates ksum.
// Block = 256 thr (8 waves), each wave 16 of the 128 phi-rows, 5 col tiles.
// ---------------------------------------------------------------------------
__global__ __launch_bounds__(256) void kv_kernel(
    const float* __restrict__ k, const float* __restrict__ v,
    float* __restrict__ kvx) {
    __shared__ float lK[16 * 64];
    __shared__ float lV[16 * 80];
    const int bh = blockIdx.x;
    const int b = bh / H_, h = bh % H_;
    const int t = threadIdx.x, lane = t & 31, wave = t >> 5;

    // ones/zeros pad columns 64..79 (written once)
    {
        int r = t >> 4, c = t & 15;
        lV[r * 80 + 64 + c] = (c == 0) ? 1.0f : 0.0f;
    }
    v8f acc[5] = {};
    const float* kb = k + ((size_t)b * N_) * INNER_ + h * DH_;
    const float* vb = v + ((size_t)b * N_) * INNER_ + h * DH_;
    const int mrow = wave * 16 + (lane & 15);     // phi-row 0..127
    const int koff = (lane >> 4) * 2;
    const int nn   = lane & 15;

    for (int n0 = 0; n0 < N_; n0 += 16) {
        __syncthreads();
        {
            int idx = t * 4;
            int r = idx >> 6, c = idx & 63;
            *(float4*)(lK + r * 64 + c) = *(const float4*)(kb + (size_t)(n0 + r) * INNER_ + c);
            *(float4*)(lV + r * 80 + c) = *(const float4*)(vb + (size_t)(n0 + r) * INNER_ + c);
        }
        __syncthreads();
        #pragma unroll
        for (int ks = 0; ks < 4; ks++) {
            const int kk = ks * 4 + koff;
            v2f a;
            {
                float r0 = (mrow < 64) ? lK[kk * 64 + mrow] : -lK[kk * 64 + mrow - 64];
                float r1 = (mrow < 64) ? lK[(kk + 1) * 64 + mrow] : -lK[(kk + 1) * 64 + mrow - 64];
                a.x = fmaxf(r0, 0.0f);
                a.y = fmaxf(r1, 0.0f);
            }
            #pragma unroll
            for (int c = 0; c < 5; c++) {
                v2f bv;
                bv.x = lV[kk * 80 + c * 16 + nn];
                bv.y = lV[(kk + 1) * 80 + c * 16 + nn];
                acc[c] = wmma4(a, bv, acc[c]);
            }
        }
    }
    float* outp = kvx + (size_t)bh * 128 * 80;
    const int half = lane >> 4;
    #pragma unroll
    for (int c = 0; c < 5; c++)
        #pragma unroll
        for (int j = 0; j < 8; j++) {
            const int row = wave * 16 + j + half * 8;
            outp[row * 80 + c * 16 + nn] = acc[c][j];
        }
}

// ---------------------------------------------------------------------------
// attn_kernel: out[n, 0..63] = (phi(q)/sqrt(128)) @ kvx ; col 64 = normalizer.
// Block = 128 thr (4 waves), 64 n-rows per block, K = 128.
// ---------------------------------------------------------------------------
__global__ __launch_bounds__(128) void attn_kernel(
    const float* __restrict__ q,    // (G,B,N,768) post-transform
    const float* __restrict__ kvx,  // (B*H,128,80)
    float* __restrict__ att) {      // (B,N,3072) as (b,n,g,h,d)
    __shared__ float lKV[128 * 80];   // 40 KB
    __shared__ float lQ[64 * 64];     // 16 KB
    const int gbh = blockIdx.x;
    const int g = gbh / (B_ * H_);
    const int bh = gbh % (B_ * H_);
    const int b = bh / H_, h = bh % H_;
    const int nBase = blockIdx.y * 64;
    const int t = threadIdx.x, lane = t & 31, wave = t >> 5;

    const float* kvp = kvx + (size_t)bh * 128 * 80;
    for (int i = t * 4; i < 128 * 80; i += 512)
        *(float4*)(lKV + i) = *(const float4*)(kvp + i);
    const float* qb = q + (((size_t)g * B_ + b) * N_ + nBase) * INNER_ + h * DH_;
    for (int i = t * 4; i < 64 * 64; i += 512) {
        int r = i >> 6, c = i & 63;
        *(float4*)(lQ + i) = *(const float4*)(qb + (size_t)r * INNER_ + c);
    }
    __syncthreads();

    v8f acc[5] = {};
    const float invs = 0.088388347648318447f;   // 1/sqrt(2*DH)
    const int m = wave * 16 + (lane & 15);      // local n row 0..63
    const int koff = (lane >> 4) * 2;
    const int nn = lane & 15;
    #pragma unroll 4
    for (int k0 = 0; k0 < 128; k0 += 4) {
        const int kk = k0 + koff;
        v2f a;
        {
            float r0 = (kk < 64) ? lQ[m * 64 + kk] : -lQ[m * 64 + kk - 64];
            float r1 = (kk + 1 < 64) ? lQ[m * 64 + kk + 1] : -lQ[m * 64 + kk + 1 - 64];
            a.x = fmaxf(r0, 0.0f) * invs;
            a.y = fmaxf(r1, 0.0f) * invs;
        }
        #pragma unroll
        for (int c = 0; c < 5; c++) {
            v2f bv;
            bv.x = lKV[kk * 80 + c * 16 + nn];
            bv.y = lKV[(kk + 1) * 80 + c * 16 + nn];
            acc[c] = wmma4(a, bv, acc[c]);
        }
    }
    // divide by normalizer (col 64 = tile 4, nn==0, lanes 0/16) and store
    const int half = lane >> 4;
    float* ob = att + (((size_t)b * N_) + nBase) * (G_ * INNER_) + (size_t)g * INNER_ + h * DH_;
    #pragma unroll
    for (int j = 0; j < 8; j++) {
        const float nrm = __shfl(acc[4][j], lane & 16, 32);
        const float sc = 1.0f / (nrm + 1e-6f);
        const int row = wave * 16 + j + half * 8;
        #pragma unroll
        for (int c = 0; c < 4; c++)
            ob[(size_t)row * (G_ * INNER_) + c * 16 + nn] = acc[c][j] * sc;
    }
}

// ---------------------------------------------------------------------------
// LN(768) + exact GELU, in place. One block (256 thr) per row.
// ---------------------------------------------------------------------------
__global__ __launch_bounds__(256) void lngelu_kernel(
    float* __restrict__ y, const float* __restrict__ gf,
    const float* __restrict__ bef) {
    __shared__ float red[2][8];
    float* p = y + (size_t)blockIdx.x * INNER_;
    const int t = threadIdx.x;
    float v0 = p[t], v1 = p[t + 256], v2 = p[t + 512];
    float s = v0 + v1 + v2, s2 = v0 * v0 + v1 * v1 + v2 * v2;
    #pragma unroll
    for (int off = 16; off > 0; off >>= 1) {
        s  += __shfl_xor(s, off, 32);
        s2 += __shfl_xor(s2, off, 32);
    }
    const int wave = t >> 5, lane = t & 31;
    if (lane == 0) { red[0][wave] = s; red[1][wave] = s2; }
    __syncthreads();
    float stot = 0.0f, s2tot = 0.0f;
    #pragma unroll
    for (int i = 0; i < 8; i++) { stot += red[0][i]; s2tot += red[1][i]; }
    const float mean = stot * (1.0f / 768.0f);
    const float var  = s2tot * (1.0f / 768.0f) - mean * mean;
    const float inv  = rsqrtf(var + 1e-5f);
    float z0 = (v0 - mean) * inv * gf[t] + bef[t];
    float z1 = (v1 - mean) * inv * gf[t + 256] + bef[t + 256];
    float z2 = (v2 - mean) * inv * gf[t + 512] + bef[t + 512];
    p[t]       = geluf(z0);
    p[t + 256] = geluf(z1);
    p[t + 512] = geluf(z2);
}

// ---------------------------------------------------------------------------
extern "C" void kernel_launch(void* const* d_in, const int* in_sizes, int n_in,
                              void* d_out, int out_size, void* d_ws, size_t ws_size,
                              hipStream_t stream) {
    (void)in_sizes; (void)n_in; (void)out_size; (void)ws_size;
    const float* x     = (const float*)d_in[0];
    const float* Wk    = (const float*)d_in[1];
    const float* Wv    = (const float*)d_in[2];
    const float* Wq    = (const float*)d_in[3];
    const float* Wt    = (const float*)d_in[4];
    const float* bt    = (const float*)d_in[5];
    const float* gt    = (const float*)d_in[6];
    const float* betat = (const float*)d_in[7];
    const float* Wf    = (const float*)d_in[8];
    const float* bf    = (const float*)d_in[9];
    const float* gf    = (const float*)d_in[10];
    const float* betaf = (const float*)d_in[11];
    const float* Wo    = (const float*)d_in[12];
    const float* bo    = (const float*)d_in[13];
    float* out = (float*)d_out;

    float* ws = (float*)d_ws;
    const size_t BN  = (size_t)B_ * N_;        // 32768
    const size_t KSZ = BN * INNER_;            // 25,165,824
    const size_t QSZ = (size_t)G_ * KSZ;       // 100,663,296
    const size_t ATT = BN * (size_t)(G_ * INNER_);
    float* ws_q   = ws;
    float* ws_att = ws_q + QSZ;
    float* ws_k   = ws_att + ATT;
    float* ws_v   = ws_k + KSZ;
    float* ws_kv  = ws_v + KSZ;                // 96*128*80 floats
    float* ws_fus = ws_v;                      // alias: V dead after kv_kernel

    const dim3 gB(INNER_ / 64, BN / 128);      // (12, 256)
    gemm_f32_wmma<<<gB, 256, 0, stream>>>(x, Wk, nullptr, ws_k, (int)BN, INNER_, DIM_);
    gemm_f32_wmma<<<gB, 256, 0, stream>>>(x, Wv, nullptr, ws_v, (int)BN, INNER_, DIM_);
    for (int g = 0; g < G_; g++)
        gemm_f32_wmma<<<gB, 256, 0, stream>>>(x, Wq + (size_t)g * DIM_ * INNER_, nullptr,
                                              ws_q + (size_t)g * KSZ, (int)BN, INNER_, DIM_);
    qtransform_kernel<<<(G_ * B_ * N_ * H_) / 8, 256, 0, stream>>>(ws_q, Wt, bt, gt, betat);
    kv_kernel<<<B_ * H_, 256, 0, stream>>>(ws_k, ws_v, ws_kv);
    attn_kernel<<<dim3(G_ * B_ * H_, N_ / 64), 128, 0, stream>>>(ws_q, ws_kv, ws_att);
    gemm_f32_wmma<<<gB, 256, 0, stream>>>(ws_att, Wf, bf, ws_fus, (int)BN, INNER_, G_ * INNER_);
    lngelu_kernel<<<(int)BN, 256, 0, stream>>>(ws_fus, gf, betaf);
    gemm_f32_wmma<<<gB, 256, 0, stream>>>(ws_fus, Wo, bo, out, (int)BN, DIM_, INNER_);
}